// HashGridEncoding_29446295781376
// MI455X (gfx1250) — compile-verified
//
#include <hip/hip_runtime.h>
#include <cstdint>
#include <cmath>

// HashGrid encoding (Instant-NGP style) for MI455X / gfx1250.
//
// Roofline: ~0.7 GFLOP vs ~204 MB compulsory HBM traffic and ~1.07 GB of
// random 8B gathers that live entirely in the 192MB L2 (tables are 64MB).
// => latency/gather-request bound. WMMA does not apply (per-point independent
// length-8 dot products; no shared matrix operand). Optimizations:
//   * one global_load_b64 per corner (float2), halving gather requests
//   * u32 hash math (low 19 bits invariant under 64->32 truncation; all
//     operands non-negative, so % 2^19 == & 0x7FFFF)
//   * full 16-level unroll -> deep LOADcnt pipelining of gathers
//   * CDNA5 async global->LDS DMA (ASYNCcnt) to stage coords coalesced
//   * wave32-native 256-thread blocks (8 waves)

#define NUM_LEVELS 16
#define TABLE_SIZE 524288u          // 2^19
#define TMASK      (TABLE_SIZE - 1u)
#define N_POINTS   1048576
#define BLOCK      256
#define PRIME_Y    2654435761u
#define PRIME_Z    805459861u

struct LevelParams {
    float resm1f[NUM_LEVELS];
    int   resm1i[NUM_LEVELS];
};

__global__ __launch_bounds__(BLOCK)
void hashgrid_enc_kernel(const float* __restrict__ xin,
                         const float* __restrict__ tables,
                         float* __restrict__ out,
                         LevelParams lp)
{
    __shared__ float tile[3 * BLOCK];
    const int t  = threadIdx.x;
    const int n0 = blockIdx.x * BLOCK;

    // ---- Stage this block's 256 xyz triplets (768 dwords) into LDS via the
    // CDNA5 async global->LDS path (tracked by ASYNCcnt), fully coalesced. ----
    {
        const float* g = xin + (size_t)3 * (size_t)n0;
        unsigned lds0 = (unsigned)(uintptr_t)(void*)&tile[0] + 4u * (unsigned)t;
        uint64_t ga   = (uint64_t)(uintptr_t)(g + t);
        asm volatile("global_load_async_to_lds_b32 %0, %1, off"
                     :: "v"(lds0), "v"(ga) : "memory");
        asm volatile("global_load_async_to_lds_b32 %0, %1, off"
                     :: "v"(lds0 + 4u * BLOCK), "v"(ga + 4ull * BLOCK) : "memory");
        asm volatile("global_load_async_to_lds_b32 %0, %1, off"
                     :: "v"(lds0 + 8u * BLOCK), "v"(ga + 8ull * BLOCK) : "memory");
        asm volatile("s_wait_asynccnt 0x0" ::: "memory");
    }
    __syncthreads();

    // lane t reads dwords 3t..3t+2: stride-3 over 64 banks -> conflict-free
    const float px = tile[3 * t + 0];
    const float py = tile[3 * t + 1];
    const float pz = tile[3 * t + 2];

    float2* __restrict__ po =
        (float2*)(out + (size_t)(n0 + t) * (size_t)(2 * NUM_LEVELS));

#pragma unroll
    for (int l = 0; l < NUM_LEVELS; ++l) {
        const float rm1f = lp.resm1f[l];
        const int   rm1i = lp.resm1i[l];

        const float xs = px * rm1f;
        const float ys = py * rm1f;
        const float zs = pz * rm1f;

        int fx = (int)floorf(xs);
        int fy = (int)floorf(ys);
        int fz = (int)floorf(zs);
        fx = fx < 0 ? 0 : (fx > rm1i ? rm1i : fx);
        fy = fy < 0 ? 0 : (fy > rm1i ? rm1i : fy);
        fz = fz < 0 ? 0 : (fz > rm1i ? rm1i : fz);
        const int cx = (fx + 1) > rm1i ? rm1i : (fx + 1);
        const int cy = (fy + 1) > rm1i ? rm1i : (fy + 1);
        const int cz = (fz + 1) > rm1i ? rm1i : (fz + 1);

        // weights use the clipped floor, matching the reference
        const float wx = xs - (float)fx;
        const float wy = ys - (float)fy;
        const float wz = zs - (float)fz;
        const float ox = 1.0f - wx, oy = 1.0f - wy, oz = 1.0f - wz;

        // spatial hash, u32 is exact for the low 19 bits
        const unsigned hx0 = (unsigned)fx;
        const unsigned hx1 = (unsigned)cx;
        const unsigned hy0 = (unsigned)fy * PRIME_Y;
        const unsigned hy1 = (unsigned)cy * PRIME_Y;
        const unsigned hz0 = (unsigned)fz * PRIME_Z;
        const unsigned hz1 = (unsigned)cz * PRIME_Z;

        const unsigned i000 = (hx0 ^ hy0 ^ hz0) & TMASK;
        const unsigned i100 = (hx1 ^ hy0 ^ hz0) & TMASK;
        const unsigned i010 = (hx0 ^ hy1 ^ hz0) & TMASK;
        const unsigned i110 = (hx1 ^ hy1 ^ hz0) & TMASK;
        const unsigned i001 = (hx0 ^ hy0 ^ hz1) & TMASK;
        const unsigned i101 = (hx1 ^ hy0 ^ hz1) & TMASK;
        const unsigned i011 = (hx0 ^ hy1 ^ hz1) & TMASK;
        const unsigned i111 = (hx1 ^ hy1 ^ hz1) & TMASK;

        const float2* __restrict__ tb =
            (const float2*)tables + (size_t)l * (size_t)TABLE_SIZE;

        // 8 x global_load_b64 gathers, all in flight together
        const float2 f000 = tb[i000];
        const float2 f100 = tb[i100];
        const float2 f010 = tb[i010];
        const float2 f110 = tb[i110];
        const float2 f001 = tb[i001];
        const float2 f101 = tb[i101];
        const float2 f011 = tb[i011];
        const float2 f111 = tb[i111];

        const float a00 = ox * oy, a10 = wx * oy, a01 = ox * wy, a11 = wx * wy;
        const float w000 = a00 * oz, w100 = a10 * oz, w010 = a01 * oz, w110 = a11 * oz;
        const float w001 = a00 * wz, w101 = a10 * wz, w011 = a01 * wz, w111 = a11 * wz;

        float r0 = f000.x * w000;
        float r1 = f000.y * w000;
        r0 = fmaf(f100.x, w100, r0);  r1 = fmaf(f100.y, w100, r1);
        r0 = fmaf(f010.x, w010, r0);  r1 = fmaf(f010.y, w010, r1);
        r0 = fmaf(f110.x, w110, r0);  r1 = fmaf(f110.y, w110, r1);
        r0 = fmaf(f001.x, w001, r0);  r1 = fmaf(f001.y, w001, r1);
        r0 = fmaf(f101.x, w101, r0);  r1 = fmaf(f101.y, w101, r1);
        r0 = fmaf(f011.x, w011, r0);  r1 = fmaf(f011.y, w011, r1);
        r0 = fmaf(f111.x, w111, r0);  r1 = fmaf(f111.y, w111, r1);

        po[l] = make_float2(r0, r1);   // 8B-aligned b64 store
    }
}

extern "C" void kernel_launch(void* const* d_in, const int* in_sizes, int n_in,
                              void* d_out, int out_size, void* d_ws, size_t ws_size,
                              hipStream_t stream) {
    (void)in_sizes; (void)n_in; (void)out_size; (void)d_ws; (void)ws_size;

    // Reproduce the reference's resolution table bit-exactly on the host:
    // numpy scalar log/exp/** route through the system libm, so evaluating the
    // identical double-precision expression here matches np.floor()'s result,
    // including the razor-edge level-15 value (16 * g^15 ~ 2048 +/- 2e-12).
    LevelParams lp;
    const double growth = exp((log(2048.0) - log(16.0)) / 15.0);
    for (int l = 0; l < NUM_LEVELS; ++l) {
        const int r = (int)floor(16.0 * pow(growth, (double)l));
        lp.resm1f[l] = (float)(r - 1);
        lp.resm1i[l] = r - 1;
    }

    const float* x      = (const float*)d_in[0];
    const float* tables = (const float*)d_in[1];
    float*       out    = (float*)d_out;

    hashgrid_enc_kernel<<<dim3(N_POINTS / BLOCK), dim3(BLOCK), 0, stream>>>(
        x, tables, out, lp);
}